// PartiallyUnsharedConv1d_46299747451574
// MI455X (gfx1250) — compile-verified
//
#include <hip/hip_runtime.h>

// ---------------------------------------------------------------------------
// PartiallyUnsharedConv1d on MI455X (gfx1250, wave32, WMMA)
//
// out[b,dk,o,l] = bias[dk,o,chunk(l)] + sum_{i,f} W[dk,o,i,chunk(l),f] * x[b,dk,i,l+f-4]
//
// GEMM per (b,dk,chunk-block): M=64(o), N=128(l), K=576 (= f*64+i), fp32,
// v_wmma_f32_16x16x4_f32. LDS: x tile first (pair-interleaved rows, so each
// B fragment is one ds_load_b64 with an immediate offset), then the 64x576
// weight tile (row stride 580 -> conflict-free b64 A reads). All per-step DS
// offsets fit the 16-bit immediate. Staging via global_load_async_to_lds_b32.
// ---------------------------------------------------------------------------

typedef __attribute__((ext_vector_type(2))) float v2f;
typedef __attribute__((ext_vector_type(8))) float v8f;

#define SXP 288      // Xs pair-row stride in dwords (32 pair-rows, cols 0..135 x2)
#define XS_DW (32 * SXP)            // 9216 dwords = 36,864 B (LDS offset 0)
#define SA 580       // As row stride in dwords (64 o rows, K = f*64+i, K<576)
#define LBLK 128

#if __has_builtin(__builtin_amdgcn_global_load_async_to_lds_b32)
#define HAS_ASYNC 1
#endif
#if __has_builtin(__builtin_amdgcn_wmma_f32_16x16x4_f32)
#define HAS_WMMA 1
#endif

typedef __attribute__((address_space(1))) int gint_t;  // global int*
typedef __attribute__((address_space(3))) int lint_t;  // LDS int*

__device__ __forceinline__ void stage32(const float* g, float* l) {
#ifdef HAS_ASYNC
  __builtin_amdgcn_global_load_async_to_lds_b32((gint_t*)g, (lint_t*)l, 0, 0);
#else
  *l = *g;
#endif
}

__device__ __forceinline__ v8f wmma_f32x4(v2f a, v2f b, v8f c) {
#ifdef HAS_WMMA
  return __builtin_amdgcn_wmma_f32_16x16x4_f32(
      /*neg_a=*/false, a, /*neg_b=*/false, b,
      /*c_mod=*/(short)0, c, /*reuse_a=*/false, /*reuse_b=*/false);
#else
  c[0] = fmaf(a.x, b.x, fmaf(a.y, b.y, c[0]));
  return c;
#endif
}

__global__ __launch_bounds__(512) void puconv1d_wmma_kernel(
    const float* __restrict__ x,     // (8,2,4,64,4096)
    const float* __restrict__ w,     // (2,4,64,64,8,1,9)
    const float* __restrict__ bias,  // (2,4,64,8,1)
    float* __restrict__ out)         // (8,2,4,64,4096)
{
  extern __shared__ float smem[];
  float* Xs = smem;            // XS_DW floats: x tile, pair-interleaved rows
  float* As = smem + XS_DW;    // 64*SA floats: weights, K-order f*64+i

  const int tid = threadIdx.x;
  const int dk  = blockIdx.y;   // 0..7  (= d*4+k)
  const int b   = blockIdx.z;   // 0..7

  // ---- map blockIdx.x (0..32) -> (chunk, block-in-chunk, l0, len) ----------
  int bx = blockIdx.x;
  int chunk = 0, base = 0, nb = 4;               // chunks 0..6: 505 -> 4 blocks
  while (bx >= base + nb) {                      // chunk 7: 561 -> 5 blocks
    base += nb;
    ++chunk;
    nb = (chunk == 7) ? 5 : 4;
  }
  const int blk    = bx - base;
  const int clen   = (chunk == 7) ? 561 : 505;
  const int l0     = chunk * 505 + blk * LBLK;
  const int len    = min(LBLK, clen - blk * LBLK);

  // ---- stage weights: w[dk,o,i,chunk,0,f] -> As[o*SA + f*64 + i] -----------
  {
    const float* wbase = w + (size_t)dk * (64 * 64 * 8 * 9) + chunk * 9;
#pragma unroll 4
    for (int it = 0; it < 72; ++it) {
      int e    = tid + it * 512;      // e < 36864
      int f    = e % 9;
      int rest = e / 9;               // o*64 + i
      int i    = rest & 63;
      int o    = rest >> 6;
      stage32(wbase + (size_t)(o * 64 + i) * 72 + f, &As[o * SA + f * 64 + i]);
    }
  }

  // ---- stage x tile: x[b,dk,i, l0-4 .. l0+131] ----------------------------
  // pair-interleaved: element (i,col) -> Xs[(i>>1)*SXP + col*2 + (i&1)]
  {
    const float* xbase = x + (size_t)(b * 8 + dk) * 64 * 4096;
#pragma unroll 4
    for (int it = 0; it < 17; ++it) {
      int e   = tid + it * 512;       // e < 8704 = 64*136
      int row = e / 136;
      int col = e - row * 136;
      int l   = l0 - 4 + col;
      float* dst = &Xs[(row >> 1) * SXP + col * 2 + (row & 1)];
      if ((unsigned)l < 4096u) {
        stage32(xbase + (size_t)row * 4096 + l, dst);
      } else {
        *dst = 0.0f;
      }
    }
  }

#ifdef HAS_ASYNC
  asm volatile("s_wait_asynccnt 0" ::: "memory");   // drain ASYNCcnt
#endif
  __syncthreads();

  // ---- per-wave tile assignment -------------------------------------------
  const int wv   = tid >> 5;       // 16 waves
  const int lane = tid & 31;
  const int hi   = lane >> 4;      // lane half (0/1)
  const int n    = lane & 15;      // N within 16-wide tile
  const int ot   = wv & 3;         // o-tile 0..3 (16 rows each)
  const int lp   = wv >> 2;        // l-pair 0..3 -> l offsets lp*32 + {0,16}

  // A fragment: lane holds A[o = ot*16+n][K = 4s + 2*hi + {0,1}]  (ds_load_b64)
  const int aBase = (ot * 16 + n) * SA + 2 * hi;
  // B fragment: lane holds X[i = 4s + 2*hi + {0,1}][col = lp*32 + t*16 + n + f]
  //   pair = 2s + hi, address = pair*SXP + col*2 + v  -> one ds_load_b64
  const int bBase = hi * SXP + (lp * 32 + n) * 2;

  // ---- accumulators initialized with bias ---------------------------------
  const float* bb = bias + (size_t)dk * 64 * 8 + chunk;
  v8f acc0, acc1;
#pragma unroll
  for (int r = 0; r < 8; ++r) {
    float bv = bb[(ot * 16 + r + 8 * hi) * 8];
    acc0[r] = bv;
    acc1[r] = bv;
  }

  // ---- main K loop: 9 taps x 16 sub-steps of K=4 --------------------------
#pragma unroll 1
  for (int f = 0; f < 9; ++f) {
    const float* ap  = &As[aBase + f * 64];
    const float* bp0 = &Xs[bBase + 2 * f];
    const float* bp1 = bp0 + 32;
#pragma unroll
    for (int s = 0; s < 16; ++s) {
      v2f a  = *(const v2f*)&ap[s * 4];            // imm offset s*16 (+As base)
      v2f b0 = *(const v2f*)&bp0[s * (2 * SXP)];   // imm offset s*2304
      v2f b1 = *(const v2f*)&bp1[s * (2 * SXP)];
      acc0 = wmma_f32x4(a, b0, acc0);
      acc1 = wmma_f32x4(a, b1, acc1);
    }
  }

  // ---- store: out[b,dk,o, l0 + lp*32 + t*16 + n], guarded by chunk len ----
  float* obase = out + (size_t)(b * 8 + dk) * 64 * 4096;
#pragma unroll
  for (int r = 0; r < 8; ++r) {
    int o = ot * 16 + r + 8 * hi;
    float* orow = obase + (size_t)o * 4096 + l0;
    int c0 = lp * 32 + n;
    if (c0 < len)      orow[c0]      = acc0[r];
    if (c0 + 16 < len) orow[c0 + 16] = acc1[r];
  }
}

extern "C" void kernel_launch(void* const* d_in, const int* in_sizes, int n_in,
                              void* d_out, int out_size, void* d_ws, size_t ws_size,
                              hipStream_t stream) {
  const float* x    = (const float*)d_in[0];
  const float* w    = (const float*)d_in[1];
  const float* bias = (const float*)d_in[2];
  float* out        = (float*)d_out;

  dim3 grid(33, 8, 8);     // 33 l-blocks x 8 (d,k) groups x 8 batches
  dim3 block(512, 1, 1);   // 16 wave32 waves
  size_t shmem = (size_t)(XS_DW + 64 * SA) * sizeof(float);  // 185,344 B

  hipLaunchKernelGGL(puconv1d_wmma_kernel, grid, block, shmem, stream,
                     x, w, bias, out);
}